// BaseSO3Convolution_17248588661207
// MI455X (gfx1250) — compile-verified
//
#include <hip/hip_runtime.h>

// ---------------------------------------------------------------------------
// SO(3) convolution (SchNetPack-style), lmax=2, C=128, fused for MI455X.
//   Pass 1: Wpre[E,384] = (radial[E,20] @ Wf[20,384] + bf) * cutoff  (WMMA f32)
//   Pass 2: per-edge CG contraction, gather x[idx_j], atomic scatter y[idx_i]
// ---------------------------------------------------------------------------

#define SH 9
#define NB 128
#define NR 20
#define LW 384   // (LMAX+1)*NB

typedef __attribute__((ext_vector_type(2))) float v2f;
typedef __attribute__((ext_vector_type(8))) float v8f;

// ---- hand-derived real-SH Clebsch-Gordan nonzeros for lmax = 2 ------------
// SH index order: 0:(0,0) 1:(1,-1)=y 2:(1,0)=z 3:(1,+1)=x
//                 4:(2,-2)=xy 5:(2,-1)=yz 6:(2,0)=z2 7:(2,+1)=xz 8:(2,+2)=x2y2
#define NPATH 83
__device__ constexpr int P_S1[NPATH] = {           // dir index (IDX_IN_1)
  0,0,0,0,0,0,0,0,0,                               // (0,l,l)
  1,2,3,4,5,6,7,8,                                 // (l,0,l)
  1,2,3,                                           // (1,1,0)
  1,1,3,3,2,2,3,2,1,3,1,                           // (1,1,2)
  2,2,2,1,3,1,3,1,3,1,3,                           // (1,2,1)
  6,5,7,6,6,5,7,4,4,8,8,                           // (2,1,1)
  4,5,6,7,8,                                       // (2,2,0)
  6,6,7,6,5,6,4,6,8,7,7,5,5,7,5,4,5,4,7,8,5,8,7,4,8 // (2,2,2)
};
__device__ constexpr int P_S2[NPATH] = {           // x-gather index (IDX_IN_2)
  0,1,2,3,4,5,6,7,8,
  0,0,0,0,0,0,0,0,
  1,2,3,
  1,1,3,3,2,3,2,1,2,1,3,
  6,5,7,6,6,5,7,4,4,8,8,
  2,2,2,1,3,1,3,1,3,1,3,
  4,5,6,7,8,
  6,7,6,5,6,4,6,8,6,7,7,5,5,5,7,5,7,7,4,5,8,7,8,4,8
};
__device__ constexpr int P_O[NPATH] = {            // output index (IDX_OUT)
  0,1,2,3,4,5,6,7,8,
  1,2,3,4,5,6,7,8,
  0,0,0,
  6,8,6,8,6,7,7,5,5,4,4,
  2,1,3,1,3,2,2,3,1,1,3,
  2,1,3,1,3,2,2,3,1,1,3,
  0,0,0,0,0,
  6,7,7,5,5,4,4,8,8,6,8,6,8,4,4,7,7,5,5,5,5,7,7,6,6
};
__device__ constexpr float P_CG[NPATH] = {
  1.f,1.f,1.f,1.f,1.f,1.f,1.f,1.f,1.f,
  1.f,1.f,1.f,1.f,1.f,1.f,1.f,1.f,
  -0.57735027f,-0.57735027f,-0.57735027f,
  -0.40824829f,-0.70710678f,-0.40824829f, 0.70710678f, 0.81649658f,
   0.70710678f, 0.70710678f, 0.70710678f, 0.70710678f, 0.70710678f, 0.70710678f,
  -0.63245553f,-0.54772256f,-0.54772256f, 0.31622777f, 0.31622777f,
  -0.54772256f,-0.54772256f,-0.54772256f,-0.54772256f, 0.54772256f,-0.54772256f,
  -0.63245553f,-0.54772256f,-0.54772256f, 0.31622777f, 0.31622777f,
  -0.54772256f,-0.54772256f,-0.54772256f,-0.54772256f, 0.54772256f,-0.54772256f,
   0.44721360f, 0.44721360f, 0.44721360f, 0.44721360f, 0.44721360f,
  -0.53452248f,-0.26726124f,-0.26726124f,-0.26726124f,-0.26726124f,
   0.53452248f, 0.53452248f, 0.53452248f, 0.53452248f,
  -0.26726124f,-0.46291005f,-0.26726124f, 0.46291005f,
  -0.46291005f,-0.46291005f,-0.46291005f,-0.46291005f,-0.46291005f,-0.46291005f,
   0.46291005f, 0.46291005f,-0.46291005f,-0.46291005f, 0.53452248f, 0.53452248f
};
// filter degree per path: l(s1)
__device__ constexpr int P_L[NPATH] = {
  0,0,0,0,0,0,0,0,0,
  1,1,1,2,2,2,2,2,
  1,1,1,
  1,1,1,1,1,1,1,1,1,1,1,
  1,1,1,1,1,1,1,1,1,1,1,
  2,2,2,2,2,2,2,2,2,2,2,
  2,2,2,2,2,
  2,2,2,2,2,2,2,2,2,2,2,2,2,2,2,2,2,2,2,2,2,2,2,2,2
};

// ---------------------------------------------------------------------------
// Kernel 1: dense radial filter via V_WMMA_F32_16X16X4_F32.
// One wave per 16-edge x 16-output tile; K=20 as 5 steps of K=4.
// A (16x4, f32): lanes 0-15 hold M rows; VGPR0={K0 | K2(hi lanes)}, VGPR1={K1|K3}
// B (4x16, f32): lanes 0-15 N cols; VGPR0={K0 | K2}, VGPR1={K1 | K3}
// C/D (16x16):   VGPR v -> row v (lanes 0-15) / row v+8 (lanes 16-31)
// Grid: x = N tiles (24), y = M tiles (E/16).
// ---------------------------------------------------------------------------
__global__ __launch_bounds__(32) void so3_dense_wmma(
    const float* __restrict__ radial,  // [E,20]
    const float* __restrict__ Wf,      // [20,384]
    const float* __restrict__ bf,      // [384]
    const float* __restrict__ cutoff,  // [E,1]
    float* __restrict__ Wpre,          // [E,384]
    int E)
{
  const int e0 = blockIdx.y * 16;
  const int n0 = blockIdx.x * 16;
  const int lane = threadIdx.x & 31;
  const int half = lane >> 4;                  // 0: K lo pair, 1: K hi pair
  const int l15  = lane & 15;

  v8f acc = {};
#pragma unroll
  for (int k0 = 0; k0 < NR; k0 += 4) {
    const int ka = k0 + 2 * half;
    const int er = min(e0 + l15, E - 1);       // A row (edge), clamped tail
    v2f a, b;
    a.x = radial[er * NR + ka + 0];
    a.y = radial[er * NR + ka + 1];
    b.x = Wf[(ka + 0) * LW + n0 + l15];
    b.y = Wf[(ka + 1) * LW + n0 + l15];
    acc = __builtin_amdgcn_wmma_f32_16x16x4_f32(
        /*neg_a=*/false, a, /*neg_b=*/false, b,
        /*c_mod=*/(short)0, acc, /*reuse_a=*/false, /*reuse_b=*/false);
  }

  const float bias = bf[n0 + l15];
  if (e0 + 16 <= E) {
    // interior tile: uniform branch, no per-row exec games
#pragma unroll
    for (int v = 0; v < 8; ++v) {
      const int e = e0 + v + 8 * half;         // C/D row layout
      Wpre[(size_t)e * LW + n0 + l15] = (acc[v] + bias) * cutoff[e];
    }
  } else {
#pragma unroll
    for (int v = 0; v < 8; ++v) {
      const int e = e0 + v + 8 * half;
      if (e < E)
        Wpre[(size_t)e * LW + n0 + l15] = (acc[v] + bias) * cutoff[e];
    }
  }
}

// ---------------------------------------------------------------------------
// Kernel 2: per-edge CG contraction. One wave32 per edge; each lane owns 4
// channels (float4 -> global_load_b128). The edge id is forced into an SGPR
// via readfirstlane so idx/dir loads become scalar (SMEM) loads and the
// per-path factor f = CG*dir[s1] is computed on the scalar pipe.
// ---------------------------------------------------------------------------
__global__ __launch_bounds__(256) void so3_gather_scatter(
    const float* __restrict__ x,       // [A,9,128]
    const float* __restrict__ Wpre,    // [E,384]
    const float* __restrict__ dir,     // [E,9]
    const int*   __restrict__ idx_i,   // [E]
    const int*   __restrict__ idx_j,   // [E]
    float* __restrict__ y,             // [A,9,128]
    int E)
{
  const int wv = (int)((blockIdx.x * (unsigned)blockDim.x + threadIdx.x) >> 5);
  const int lane = threadIdx.x & 31;
  if (wv >= E) return;
  const int e = __builtin_amdgcn_readfirstlane(wv);  // wave-uniform -> SGPR

  const int aj = idx_j[e];                     // scalar loads
  const int ai = idx_i[e];

  float d[SH];
#pragma unroll
  for (int s = 0; s < SH; ++s) d[s] = dir[e * SH + s];

  // gather neighbor features: 9 rows of 128 floats, 4 per lane (b128 loads)
  const float4* xr = reinterpret_cast<const float4*>(x + (size_t)aj * SH * NB);
  float4 xj[SH];
#pragma unroll
  for (int s = 0; s < SH; ++s) xj[s] = xr[s * 32 + lane];

  // radial filters for this edge: 3 degrees x 128 channels
  const float4* wr = reinterpret_cast<const float4*>(Wpre + (size_t)e * LW);
  float4 W[3];
#pragma unroll
  for (int l = 0; l < 3; ++l) W[l] = wr[l * 32 + lane];

  float4 acc[SH];
#pragma unroll
  for (int s = 0; s < SH; ++s) acc[s] = make_float4(0.f, 0.f, 0.f, 0.f);

#pragma unroll
  for (int p = 0; p < NPATH; ++p) {
    const float f = P_CG[p] * d[P_S1[p]];      // scalar (SGPR) factor
    const float4 wl = W[P_L[p]];
    const float4 xv = xj[P_S2[p]];
    // prod = W[l]*xj[s2] is CSE'd across paths sharing (l, s2)
    acc[P_O[p]].x = fmaf(f, wl.x * xv.x, acc[P_O[p]].x);
    acc[P_O[p]].y = fmaf(f, wl.y * xv.y, acc[P_O[p]].y);
    acc[P_O[p]].z = fmaf(f, wl.z * xv.z, acc[P_O[p]].z);
    acc[P_O[p]].w = fmaf(f, wl.w * xv.w, acc[P_O[p]].w);
  }

  float* yo = y + (size_t)ai * SH * NB;        // scalar base + lane offset
#pragma unroll
  for (int s = 0; s < SH; ++s) {
    float* p = yo + s * NB + lane * 4;
    unsafeAtomicAdd(p + 0, acc[s].x);
    unsafeAtomicAdd(p + 1, acc[s].y);
    unsafeAtomicAdd(p + 2, acc[s].z);
    unsafeAtomicAdd(p + 3, acc[s].w);
  }
}

// ---------------------------------------------------------------------------
extern "C" void kernel_launch(void* const* d_in, const int* in_sizes, int n_in,
                              void* d_out, int out_size, void* d_ws, size_t ws_size,
                              hipStream_t stream) {
  const float* x       = (const float*)d_in[0];
  const float* radial  = (const float*)d_in[1];
  const float* dir     = (const float*)d_in[2];
  const float* cutoff  = (const float*)d_in[3];
  const float* Wf      = (const float*)d_in[4];
  const float* bf      = (const float*)d_in[5];
  const int*   idx_i   = (const int*)d_in[6];
  const int*   idx_j   = (const int*)d_in[7];

  const int E = in_sizes[1] / NR;              // 10000

  // y must be zero before atomic accumulation (graph-capturable memset)
  hipMemsetAsync(d_out, 0, (size_t)out_size * sizeof(float), stream);

  float* Wpre = (float*)d_ws;                  // E*384 floats (~15.4 MB)

  const int blocksM = (E + 15) / 16;
  so3_dense_wmma<<<dim3(LW / 16, blocksM), dim3(32), 0, stream>>>(
      radial, Wf, bf, cutoff, Wpre, E);

  const int blocksB = (E + 7) / 8;             // 8 waves (edges) per block
  so3_gather_scatter<<<dim3(blocksB), dim3(256), 0, stream>>>(
      x, Wpre, dir, idx_i, idx_j, (float*)d_out, E);
}